// StateVariableFilter_45423574123132
// MI455X (gfx1250) — compile-verified
//
#include <hip/hip_runtime.h>
#include <math.h>
#include <stdint.h>

// ---------------- problem constants ----------------
#define B_    32
#define C_    2
#define L_    262144
#define K_    4
#define SEQ_  (B_*C_)          // 64 independent sequences
#define CHUNK 1024             // samples per lane-chunk
#define NCH   (L_/CHUNK)       // 256 chunks per sequence
#define TILE  32               // samples per lane per LDS tile
#define NTILE (CHUNK/TILE)     // 32 tiles per chunk
#define WPAD  33               // LDS row stride (TDM pad: +1 DWORD per 32)

#define HALF_PI_F    1.5707963267948966f
#define TWOR_SCALE_F 1.4426950408889634f

// ---------------- workspace layout (floats) ----------------
#define OFF_COEF 0                         // 32*4*5   = 640
#define OFF_A    640                       // 32*8*8   = 2048   (cascade transition A_b)
#define OFF_P    (640+2048)                // 32*8*8   = 2048   (P_b = A_b^CHUNK)
#define OFF_V    (640+2048+2048)           // SEQ*NCH*8 = 131072 (zero-init chunk end states)
#define OFF_S    (OFF_V + SEQ_*NCH*8)      // SEQ*NCH*8 = 131072 (true chunk initial states)

typedef float    v2f __attribute__((ext_vector_type(2)));
typedef float    v8f __attribute__((ext_vector_type(8)));
typedef uint32_t v4u __attribute__((ext_vector_type(4)));
typedef uint32_t v8u __attribute__((ext_vector_type(8)));

// ---------------- TDM helpers ----------------
// Issue a 2D tensor-tile load: 32 rows x 32 f32 cols, row stride CHUNK elems,
// padded in LDS by 1 DWORD per 32 DWORDs (-> effective row stride WPAD=33).
// D# group0: {count=1 | lds_addr | global_addr | type=2}
// D# group1: {data_size=4B, pad_enable, pad_interval=32DW, pad_amount=1DW,
//             tensor_dim0=32, tensor_dim1=32, tile_dim0=32, tile_dim1=32,
//             tensor_dim0_stride=CHUNK}
__device__ __forceinline__ void tdm_issue_tile(const float* gptr, uint32_t ldsOff) {
  uint64_t ga = (uint64_t)(uintptr_t)gptr;
  uint32_t w1 = __builtin_amdgcn_readfirstlane(ldsOff);
  uint32_t w2 = __builtin_amdgcn_readfirstlane((uint32_t)ga);
  uint32_t w3 = __builtin_amdgcn_readfirstlane(
                  ((uint32_t)(ga >> 32) & 0x01FFFFFFu) | (2u << 30));
  v4u g0;
  g0.x = 1u;        // count = 1 valid descriptor
  g0.y = w1;        // lds_addr (bytes, workgroup-relative)
  g0.z = w2;        // global_addr[31:0]
  g0.w = w3;        // global_addr[56:32] | type=2
  v8u g1;
  g1[0] = (2u << 16)   // data_size = 4 bytes
        | (1u << 20)   // pad_enable
        | (4u << 22);  // pad_interval: 4 -> every 32 DWORDs (pad_amount=0 -> 1 DWORD)
  g1[1] = ((uint32_t)TILE) << 16;  // tensor_dim0[15:0] at bits 63:48
  g1[2] = ((uint32_t)TILE) << 16;  // tensor_dim1[15:0] at bits 95:80
  g1[3] = ((uint32_t)TILE) << 16;  // tile_dim0 at bits 127:112
  g1[4] = (uint32_t)TILE;          // tile_dim1 = 32 rows (tile_dim2 = 0)
  g1[5] = (uint32_t)CHUNK;         // tensor_dim0_stride[31:0]
  g1[6] = 0u;
  g1[7] = 0u;
  // s_wait_dscnt 0 first: the TDM must not overwrite LDS that this wave's
  // in-flight ds_loads are still reading.
  asm volatile("s_wait_dscnt 0x0\n\t"
               "tensor_load_to_lds %0, %1"
               :: "s"(g0), "s"(g1) : "memory");
}
#define TDM_WAIT_ALL()  asm volatile("s_wait_tensorcnt 0x0" ::: "memory")
#define TDM_WAIT_ONE()  asm volatile("s_wait_tensorcnt 0x1" ::: "memory")
#define LDS_WAIT()      asm volatile("s_wait_dscnt 0x0"     ::: "memory")

// ============================================================
// Kernel A: coefficients + 8x8 cascade transition matrix A_b
// ============================================================
__global__ void svf_coef_kernel(const float* __restrict__ twoR,
                                const float* __restrict__ G,
                                const float* __restrict__ c_hp,
                                const float* __restrict__ c_bp,
                                const float* __restrict__ c_lp,
                                float* __restrict__ ws) {
  int tid = threadIdx.x;
  float* coef = ws + OFF_COEF;
  if (tid < B_*K_) {
    float gx = G[tid];
    float sig = 1.0f / (1.0f + expf(-gx));
    float g = tanf(HALF_PI_F * sig);
    float tr = twoR[tid];
    float sp = (tr > 20.0f) ? tr : log1pf(expf(tr));
    float twoRv = TWOR_SCALE_F * sp + 0.01f;
    float chp = c_hp[tid], cbp = c_bp[tid], clp = c_lp[tid];
    float G2 = g * g;
    float b0 = chp + cbp * g + clp * G2;
    float b1 = -2.0f * chp + 2.0f * clp * G2;
    float b2 = chp - cbp * g + clp * G2;
    float a0 = 1.0f + G2 + twoRv * g;
    float a1 = 2.0f * G2 - 2.0f;
    float a2 = 1.0f + G2 - twoRv * g;
    float inv = 1.0f / a0;
    coef[tid*5 + 0] = b0 * inv;
    coef[tid*5 + 1] = b1 * inv;
    coef[tid*5 + 2] = b2 * inv;
    coef[tid*5 + 3] = -(a1 * inv);
    coef[tid*5 + 4] = -(a2 * inv);
  }
  __syncthreads();
  if (tid < B_) {
    float r[8];
    for (int j = 0; j < 8; ++j) r[j] = 0.0f;
    float* A = ws + OFF_A + tid*64;
    for (int k = 0; k < K_; ++k) {
      const float* cf = coef + (tid*K_ + k)*5;
      float b0 = cf[0], b1 = cf[1], b2 = cf[2], na1 = cf[3], na2 = cf[4];
      float yrow[8];
      for (int j = 0; j < 8; ++j) {
        float e1 = (j == 2*k)     ? 1.0f : 0.0f;
        float e2 = (j == 2*k + 1) ? 1.0f : 0.0f;
        yrow[j]        = b0 * r[j] + e1;
        A[(2*k  )*8+j] = b1 * r[j] + e2 + na1 * yrow[j];
        A[(2*k+1)*8+j] = b2 * r[j]      + na2 * yrow[j];
      }
      for (int j = 0; j < 8; ++j) r[j] = yrow[j];
    }
  }
}

// ============================================================
// Kernel B: P_b = A_b^CHUNK via 10 WMMA matrix squarings
// (V_WMMA_F32_16X16X4_F32, K sliced 4-wide, accumulator-chained)
// ============================================================
__global__ void svf_power_kernel(float* __restrict__ ws) {
  __shared__ float M[16*17];
  const int lane = threadIdx.x;          // one wave32: EXEC all 1s
  const int b = blockIdx.x;
  const float* A = ws + OFF_A + b*64;
  for (int e = lane; e < 256; e += 32) {
    int r = e >> 4, c = e & 15;
    M[r*17 + c] = (r < 8 && c < 8) ? A[r*8 + c] : 0.0f;
  }
  __syncthreads();
  const bool lo  = lane < 16;
  const int  row = lane & 15;
  for (int it = 0; it < 10; ++it) {
    v8f acc = {0.f,0.f,0.f,0.f,0.f,0.f,0.f,0.f};
#pragma unroll
    for (int ks = 0; ks < 4; ++ks) {
      int ka = 4*ks + (lo ? 0 : 2);
      v2f Af, Bf;
      Af.x = M[row*17 + ka];     Af.y = M[row*17 + ka + 1];
      Bf.x = M[ka*17 + row];     Bf.y = M[(ka + 1)*17 + row];
      acc = __builtin_amdgcn_wmma_f32_16x16x4_f32(
          false, Af, false, Bf, (short)0, acc, false, false);
    }
    __syncthreads();
#pragma unroll
    for (int r2 = 0; r2 < 8; ++r2)
      M[(lo ? r2 : r2 + 8)*17 + row] = acc[r2];
    __syncthreads();
  }
  float* P = ws + OFF_P + b*64;
  for (int e = lane; e < 64; e += 32)
    P[e] = M[(e >> 3)*17 + (e & 7)];
}

// ============================================================
// Kernel C: pass A — each lane runs its 1024-sample chunk with zero
// init; input tiles are DMA'd into padded LDS by the TDM, double-
// buffered on TENSORcnt. No block barriers: each wave owns its slot.
// Grid: 128 blocks x 128 threads (4 waves).
// ============================================================
__global__ void svf_chunkstate_kernel(const float* __restrict__ x,
                                      float* __restrict__ ws) {
  __shared__ float tile[4][2][TILE*WPAD];
  const int seq     = blockIdx.x >> 1;
  const int half    = blockIdx.x & 1;
  const int warp    = threadIdx.x >> 5;
  const int lane    = threadIdx.x & 31;
  const int wavePos = half*4 + warp;     // 0..7 within sequence
  const int b       = seq >> 1;

  const float* cf = ws + OFF_COEF + b*K_*5;
  float cb0[4], cb1[4], cb2[4], cna1[4], cna2[4];
#pragma unroll
  for (int k = 0; k < 4; ++k) {
    cb0[k] = cf[k*5]; cb1[k] = cf[k*5+1]; cb2[k] = cf[k*5+2];
    cna1[k] = cf[k*5+3]; cna2[k] = cf[k*5+4];
  }
  float s1[4] = {0,0,0,0}, s2[4] = {0,0,0,0};

  const float* gbase = x + (size_t)seq*L_ + (size_t)wavePos*32*CHUNK;
  uint32_t lds0 = (uint32_t)(uintptr_t)&tile[warp][0][0];
  uint32_t lds1 = (uint32_t)(uintptr_t)&tile[warp][1][0];

  tdm_issue_tile(gbase, lds0);
  for (int t = 0; t < NTILE; ++t) {
    if (t + 1 < NTILE) {
      tdm_issue_tile(gbase + (size_t)(t+1)*TILE, (t+1)&1 ? lds1 : lds0);
      TDM_WAIT_ONE();
    } else {
      TDM_WAIT_ALL();
    }
    const float* trow = &tile[warp][t & 1][0];
#pragma unroll 8
    for (int c = 0; c < TILE; ++c) {
      float xv = trow[lane*WPAD + c];
#pragma unroll
      for (int k = 0; k < 4; ++k) {
        float y = fmaf(cb0[k], xv, s1[k]);
        s1[k] = fmaf(cb1[k], xv, fmaf(cna1[k], y, s2[k]));
        s2[k] = fmaf(cb2[k], xv, cna2[k] * y);
        xv = y;
      }
    }
  }
  const int chunk = wavePos*32 + lane;
  float* v = ws + OFF_V + ((size_t)seq*NCH + chunk)*8;
#pragma unroll
  for (int k = 0; k < 4; ++k) { v[2*k] = s1[k]; v[2*k+1] = s2[k]; }
}

// ============================================================
// Kernel D: chunk-level scan  s_{c+1} = P*s_c + v_c.
// 8 lanes per sequence, shuffle-based 8x8 matvec.
// ============================================================
__global__ void svf_scan_kernel(float* __restrict__ ws) {
  const int tid = threadIdx.x;           // 512 threads = 64 groups of 8
  const int g = tid >> 3;
  const int j = tid & 7;
  const int b = g >> 1;
  const float* Pr = ws + OFF_P + b*64 + j*8;
  float p[8];
#pragma unroll
  for (int i = 0; i < 8; ++i) p[i] = Pr[i];
  const float* v = ws + OFF_V + (size_t)g*NCH*8;
  float* sout    = ws + OFF_S + (size_t)g*NCH*8;
  const int gwbase = (tid & 31) & ~7;
  float s = 0.0f;
  for (int c = 0; c < NCH; ++c) {
    sout[c*8 + j] = s;
    float acc = v[c*8 + j];
#pragma unroll
    for (int i = 0; i < 8; ++i) {
      float si = __shfl(s, gwbase + i, 32);
      acc = fmaf(p[i], si, acc);
    }
    s = acc;
  }
}

// ============================================================
// Kernel E: pass B — rerun chunks with true initial states (input
// re-read hits the 192MB L2). TDM-loaded tiles, y written back into
// the padded tile, coalesced stores; per-wave s_wait_dscnt ordering.
// ============================================================
__global__ void svf_apply_kernel(const float* __restrict__ x,
                                 float* __restrict__ out,
                                 float* __restrict__ ws) {
  __shared__ float tile[4][2][TILE*WPAD];
  const int seq     = blockIdx.x >> 1;
  const int half    = blockIdx.x & 1;
  const int warp    = threadIdx.x >> 5;
  const int lane    = threadIdx.x & 31;
  const int wavePos = half*4 + warp;
  const int b       = seq >> 1;

  const float* cf = ws + OFF_COEF + b*K_*5;
  float cb0[4], cb1[4], cb2[4], cna1[4], cna2[4];
#pragma unroll
  for (int k = 0; k < 4; ++k) {
    cb0[k] = cf[k*5]; cb1[k] = cf[k*5+1]; cb2[k] = cf[k*5+2];
    cna1[k] = cf[k*5+3]; cna2[k] = cf[k*5+4];
  }
  const int chunk = wavePos*32 + lane;
  const float* si = ws + OFF_S + ((size_t)seq*NCH + chunk)*8;
  float s1[4], s2[4];
#pragma unroll
  for (int k = 0; k < 4; ++k) { s1[k] = si[2*k]; s2[k] = si[2*k+1]; }

  const float* gbase = x   + (size_t)seq*L_ + (size_t)wavePos*32*CHUNK;
  float*       ys    = out + (size_t)seq*L_ + (size_t)wavePos*32*CHUNK;
  uint32_t lds0 = (uint32_t)(uintptr_t)&tile[warp][0][0];
  uint32_t lds1 = (uint32_t)(uintptr_t)&tile[warp][1][0];

  tdm_issue_tile(gbase, lds0);
  for (int t = 0; t < NTILE; ++t) {
    if (t + 1 < NTILE) {
      tdm_issue_tile(gbase + (size_t)(t+1)*TILE, (t+1)&1 ? lds1 : lds0);
      TDM_WAIT_ONE();
    } else {
      TDM_WAIT_ALL();
    }
    float* trow = &tile[warp][t & 1][0];
#pragma unroll 8
    for (int c = 0; c < TILE; ++c) {
      float xv = trow[lane*WPAD + c];
#pragma unroll
      for (int k = 0; k < 4; ++k) {
        float y = fmaf(cb0[k], xv, s1[k]);
        s1[k] = fmaf(cb1[k], xv, fmaf(cna1[k], y, s2[k]));
        s2[k] = fmaf(cb2[k], xv, cna2[k] * y);
        xv = y;
      }
      trow[lane*WPAD + c] = xv;          // y back in place
    }
    LDS_WAIT();                          // wave's ds_stores visible to all lanes
#pragma unroll 4
    for (int i = 0; i < TILE; ++i)       // coalesced store: row i, col lane
      ys[(size_t)i*CHUNK + t*TILE + lane] = trow[i*WPAD + lane];
  }
}

// ============================================================
extern "C" void kernel_launch(void* const* d_in, const int* in_sizes, int n_in,
                              void* d_out, int out_size, void* d_ws, size_t ws_size,
                              hipStream_t stream) {
  const float* x    = (const float*)d_in[0];
  const float* twoR = (const float*)d_in[1];
  const float* G    = (const float*)d_in[2];
  const float* chp  = (const float*)d_in[3];
  const float* cbp  = (const float*)d_in[4];
  const float* clp  = (const float*)d_in[5];
  float* out = (float*)d_out;
  float* ws  = (float*)d_ws;              // ~1.05 MB used

  hipLaunchKernelGGL(svf_coef_kernel,       dim3(1),       dim3(128), 0, stream,
                     twoR, G, chp, cbp, clp, ws);
  hipLaunchKernelGGL(svf_power_kernel,      dim3(B_),      dim3(32),  0, stream, ws);
  hipLaunchKernelGGL(svf_chunkstate_kernel, dim3(SEQ_*2),  dim3(128), 0, stream, x, ws);
  hipLaunchKernelGGL(svf_scan_kernel,       dim3(1),       dim3(512), 0, stream, ws);
  hipLaunchKernelGGL(svf_apply_kernel,      dim3(SEQ_*2),  dim3(128), 0, stream, x, out, ws);
}